// LIFcell_metaeprop_trainee_LSG_22471268892953
// MI455X (gfx1250) — compile-verified
//
#include <hip/hip_runtime.h>
#include <stdint.h>

// ---------------------------------------------------------------------------
// LIF / PCM constants (from reference)
// ---------------------------------------------------------------------------
#define G_MAX   25.0f
#define THR     0.4f
#define DAMP    0.3f
#define NREF    5.0f
#define DECAY   0.9512294245007140f   // exp(-1/20)
#define KAPPA   0.9512294245007140f
#define LR      0.01f
#define LOG_FAC 4.3206115f            // sqrt(log((32+2.5e-7)/2.5e-7))

// Problem dims
#define BATCH 128
#define U     512
#define NIN_T 256
#define NIN_L 384
#define NOUT_T 128

typedef __attribute__((ext_vector_type(2))) float v2f;
typedef __attribute__((ext_vector_type(4))) float v4f;
typedef __attribute__((ext_vector_type(8))) float v8f;

// ---------------------------------------------------------------------------
// WMMA f32 16x16x4 (codegen-confirmed last round)
// ---------------------------------------------------------------------------
__device__ __forceinline__ v8f wmma_f32(v2f a, v2f b, v8f c) {
  return __builtin_amdgcn_wmma_f32_16x16x4_f32(false, a, false, b, (short)0, c,
                                               false, false);
}

// ---------------------------------------------------------------------------
// Deterministic per-element RNG (sample-once-per-weight semantics)
// ---------------------------------------------------------------------------
__device__ __forceinline__ uint32_t pcg(uint32_t x) {
  x = x * 747796405u + 2891336453u;
  uint32_t w = ((x >> ((x >> 28u) + 4u)) ^ x) * 277803737u;
  return (w >> 22u) ^ w;
}

// PCM read noise: noisy weight = clip(g_n/G_MAX,-1,1); noise only where g>0
__device__ __forceinline__ float noisy_w(float w, uint32_t seed) {
  float g = w * G_MAX;
  if (g > 0.0f) {
    float qs    = fminf(0.0088f * __expf(-0.65f * __logf(g)), 0.2f);
    float sigma = g * qs * LOG_FAC;
    uint32_t h1 = pcg(seed);
    uint32_t h2 = pcg(seed ^ 0x6f4f1f3bu);
    float u1 = (float)(h1 & 0xFFFFFFu) * (1.0f / 16777216.0f) + 1.0e-7f;
    float u2 = (float)(h2 & 0xFFFFFFu) * (1.0f / 16777216.0f);
    float n  = sqrtf(-2.0f * __logf(u1)) * __cosf(6.2831853f * u2);
    g = fmaxf(g + n * sigma, 0.0f);
  }
  return fminf(fmaxf(g * (1.0f / G_MAX), -1.0f), 1.0f);
}

// ---------------------------------------------------------------------------
// Kernel 0: materialize noisy weights ONCE, transposed (N x K) so GEMM B-frag
// loads become contiguous b64 per lane. Optionally zero the diagonal first
// ((1-eye)*w_rec). One thread per element; k fastest -> coalesced Wt writes.
// ---------------------------------------------------------------------------
__global__ void noise_transpose_kernel(const float* __restrict__ W,
                                       float* __restrict__ Wt,
                                       int K, int N, uint32_t seed,
                                       int zero_diag) {
  int tid = blockIdx.x * blockDim.x + threadIdx.x;
  int k = tid % K;
  int n = tid / K;
  if (n >= N) return;
  float w = W[(size_t)k * N + n];
  if (zero_diag && k == n) w = 0.0f;
  Wt[(size_t)n * K + k] = noisy_w(w, seed ^ (uint32_t)(k * N + n));
}

// ---------------------------------------------------------------------------
// Fragment loaders per ISA §7.12.2 (wave32).
// A (16x4): lanes 0-15 -> K=k,k+1 ; lanes 16-31 -> K=k+2,k+3   (b64 load)
// ---------------------------------------------------------------------------
__device__ __forceinline__ v2f load_a(const float* __restrict__ A, int K,
                                      int m0, int k, int lane) {
  int row = m0 + (lane & 15);
  int kk  = k + ((lane >> 4) << 1);
  const float* p = A + (size_t)row * K + kk;
  v2f r; r.x = p[0]; r.y = p[1];
  return r;
}

// B (4x16) from transposed noisy weights Wt[N x K]: contiguous b64 per lane
__device__ __forceinline__ v2f load_bt(const float* __restrict__ Wt, int K,
                                       int k, int n0, int lane) {
  int kk  = k + ((lane >> 4) << 1);
  int col = n0 + (lane & 15);
  const float* p = Wt + (size_t)col * K + kk;
  v2f r; r.x = p[0]; r.y = p[1];
  return r;
}

// ---------------------------------------------------------------------------
// Kernel 1/2: one LIF state step. One wave per 16x16 tile of (B x U).
// ---------------------------------------------------------------------------
__global__ void lif_state_kernel(const float* __restrict__ x, int K_in,
                                 const float* __restrict__ wt_in,
                                 const float* __restrict__ z,
                                 const float* __restrict__ wt_rec,
                                 const float* __restrict__ v,
                                 const float* __restrict__ r,
                                 float* __restrict__ nz_out,
                                 float* __restrict__ h_out) {
  int wave = (int)((blockIdx.x * blockDim.x + threadIdx.x) >> 5);
  int lane = threadIdx.x & 31;
  int mt = wave >> 5;              // 32 n-tiles across U=512
  int nt = wave & 31;
  int m0 = mt * 16, n0 = nt * 16;

  v8f acc = {};
  for (int k = 0; k < K_in; k += 4)
    acc = wmma_f32(load_a(x, K_in, m0, k, lane),
                   load_bt(wt_in, K_in, k, n0, lane), acc);
  for (int k = 0; k < U; k += 4)
    acc = wmma_f32(load_a(z, U, m0, k, lane),
                   load_bt(wt_rec, U, k, n0, lane), acc);

  // C/D layout: VGPR i -> row m0+i (lanes 0-15) / m0+8+i (lanes 16-31)
  int col   = n0 + (lane & 15);
  int rbase = m0 + ((lane >> 4) << 3);
  for (int i = 0; i < 8; ++i) {
    size_t idx = (size_t)(rbase + i) * U + col;
    float zv = z[idx];
    float rv = r[idx];
    float nv = DECAY * v[idx] + acc[i] - zv * THR;
    float sp = (nv > THR) ? 1.0f : 0.0f;
    float nz = (rv > 0.1f) ? 0.0f : sp;
    float nr = fminf(fmaxf(rv + NREF * nz - 1.0f, 0.0f), NREF);
    nz_out[idx] = nz;
    if (h_out != nullptr) {
      float h = fmaxf(DAMP * (1.0f - fabsf((nv - THR) / THR)), 0.0f) / THR;
      h_out[idx] = (nr > 0.1f) ? 0.0f : h;
    }
  }
}

// ---------------------------------------------------------------------------
// Kernel 3/4: readout  out = KAPPA*out_prev + nz @ noisy(w_out)   (K = 512)
// ---------------------------------------------------------------------------
__global__ void lif_out_kernel(const float* __restrict__ nz,
                               const float* __restrict__ wt_out,
                               const float* __restrict__ out_prev,
                               float* __restrict__ out_new, int N) {
  int wave = (int)((blockIdx.x * blockDim.x + threadIdx.x) >> 5);
  int lane = threadIdx.x & 31;
  int ntiles = N >> 4;
  int mt = wave / ntiles;
  int nt = wave % ntiles;
  int m0 = mt * 16, n0 = nt * 16;

  v8f acc = {};
  for (int k = 0; k < U; k += 4)
    acc = wmma_f32(load_a(nz, U, m0, k, lane),
                   load_bt(wt_out, U, k, n0, lane), acc);

  int col   = n0 + (lane & 15);
  int rbase = m0 + ((lane >> 4) << 3);
  for (int i = 0; i < 8; ++i) {
    size_t idx = (size_t)(rbase + i) * N + col;
    out_new[idx] = KAPPA * out_prev[idx] + acc[i];
  }
}

// ---------------------------------------------------------------------------
// Kernel 5/6: bandwidth-dominant batched outer-product update (~400 MB).
//   out[b,i,:] = w_change[b,i,:] + (LR*(DECAY*trace[b,i]+xz[b,i])) * hl[b,:]
// hl staged in LDS (ds_load_b128 once per thread); stream is read-once/
// write-once -> non-temporal b128 to keep L2 clean.
// ---------------------------------------------------------------------------
__global__ void eprop_outer_kernel(const float* __restrict__ w_change,
                                   const float* __restrict__ trace,
                                   const float* __restrict__ xz,
                                   const float* __restrict__ h,
                                   const float* __restrict__ nL,
                                   float* __restrict__ out, int I) {
  __shared__ float hl[U];
  int b = blockIdx.y;
  for (int t = threadIdx.x; t < U; t += blockDim.x)
    hl[t] = h[(size_t)b * U + t] * nL[(size_t)b * U + t];
  __syncthreads();

  int lane = threadIdx.x & 127;    // v4f index within a 512-wide row
  int sub  = threadIdx.x >> 7;     // 0/1: two rows in flight per block
  v4f hv = ((const v4f*)hl)[lane];

  const int ROWS = 16;
  int i0 = blockIdx.x * ROWS;
  for (int i = i0 + sub; i < i0 + ROWS; i += 2) {
    float tin = DECAY * trace[(size_t)b * I + i] + xz[(size_t)b * I + i];
    float c   = LR * tin;
    size_t base = ((size_t)b * I + i) * (size_t)U;
    v4f wc = __builtin_nontemporal_load((const v4f*)(w_change + base) + lane);
    v4f o  = wc + c * hv;
    __builtin_nontemporal_store(o, (v4f*)(out + base) + lane);
  }
}

// ---------------------------------------------------------------------------
// Launch. Workspace layout (floats):
//   nz_t[65536] | h[65536] | nz_l[65536] |
//   wtN_in_t[131072] wtN_rec_t[262144] wtN_out_t[65536]
//   wtN_in_l[196608] wtN_rec_l[262144] wtN_out_l[262144]      (~5.3 MB total)
// ---------------------------------------------------------------------------
extern "C" void kernel_launch(void* const* d_in, const int* in_sizes, int n_in,
                              void* d_out, int out_size, void* d_ws, size_t ws_size,
                              hipStream_t stream) {
  const float* x_t       = (const float*)d_in[0];
  const float* x_l       = (const float*)d_in[1];
  const float* v_t       = (const float*)d_in[2];
  const float* z_t       = (const float*)d_in[3];
  const float* r_t       = (const float*)d_in[4];
  const float* out_t     = (const float*)d_in[5];
  const float* v_l       = (const float*)d_in[6];
  const float* z_l       = (const float*)d_in[7];
  const float* r_l       = (const float*)d_in[8];
  const float* L         = (const float*)d_in[9];
  const float* trace_in  = (const float*)d_in[10];
  const float* trace_rec = (const float*)d_in[11];
  const float* w_in_chg  = (const float*)d_in[12];
  const float* w_rec_chg = (const float*)d_in[13];
  const float* w_in_t    = (const float*)d_in[14];
  const float* w_rec_t   = (const float*)d_in[15];
  const float* w_out_t   = (const float*)d_in[16];
  const float* w_in_l    = (const float*)d_in[17];
  const float* w_rec_l   = (const float*)d_in[18];
  const float* w_out_l   = (const float*)d_in[19];

  float* out = (float*)d_out;
  // flat output: nout_t | nw_in_change | nw_rec_change | nL
  float* nout  = out;                       // 128*128
  float* nwin  = out + 16384;               // 128*256*512
  float* nwrec = out + 16793600;            // 128*512*512
  float* nLo   = out + 50348032;            // 128*512

  float* ws    = (float*)d_ws;
  float* nz_t  = ws;
  float* h     = ws + 65536;
  float* nz_l  = ws + 131072;
  float* wtInT  = ws + 196608;              // 512 x 256
  float* wtRecT = ws + 327680;              // 512 x 512
  float* wtOutT = ws + 589824;              // 128 x 512
  float* wtInL  = ws + 655360;              // 512 x 384
  float* wtRecL = ws + 851968;              // 512 x 512
  float* wtOutL = ws + 1114112;             // 512 x 512

  // 0) one-shot noisy-weight materialization (transposed)
  noise_transpose_kernel<<<(NIN_T * U) / 256, 256, 0, stream>>>(w_in_t,  wtInT,  NIN_T, U, 0x1111u, 0);
  noise_transpose_kernel<<<(U * U)     / 256, 256, 0, stream>>>(w_rec_t, wtRecT, U, U,     0x2222u, 1);
  noise_transpose_kernel<<<(U * NOUT_T)/ 256, 256, 0, stream>>>(w_out_t, wtOutT, U, NOUT_T,0x3333u, 0);
  noise_transpose_kernel<<<(NIN_L * U) / 256, 256, 0, stream>>>(w_in_l,  wtInL,  NIN_L, U, 0x4444u, 0);
  noise_transpose_kernel<<<(U * U)     / 256, 256, 0, stream>>>(w_rec_l, wtRecL, U, U,     0x5555u, 1);
  noise_transpose_kernel<<<(U * U)     / 256, 256, 0, stream>>>(w_out_l, wtOutL, U, U,     0x6666u, 0);

  // 1) trainee state: 8x32 tiles = 256 waves, 4 waves/block
  lif_state_kernel<<<64, 128, 0, stream>>>(x_t, NIN_T, wtInT, z_t, wtRecT,
                                           v_t, r_t, nz_t, h);
  // 2) LSG state
  lif_state_kernel<<<64, 128, 0, stream>>>(x_l, NIN_L, wtInL, z_l, wtRecL,
                                           v_l, r_l, nz_l, nullptr);
  // 3) trainee readout: 8x8 tiles = 64 waves
  lif_out_kernel<<<16, 128, 0, stream>>>(nz_t, wtOutT, out_t, nout, NOUT_T);
  // 4) learning signal nL: 8x32 tiles = 256 waves
  lif_out_kernel<<<64, 128, 0, stream>>>(nz_l, wtOutL, L, nLo, U);
  // 5) nw_in_change  (64 MB each way)
  eprop_outer_kernel<<<dim3(NIN_T / 16, BATCH), 256, 0, stream>>>(
      w_in_chg, trace_in, x_t, h, nLo, nwin, NIN_T);
  // 6) nw_rec_change (128 MB each way)
  eprop_outer_kernel<<<dim3(U / 16, BATCH), 256, 0, stream>>>(
      w_rec_chg, trace_rec, z_t, h, nLo, nwrec, U);
}